// PointPillarScatter_88940182765792
// MI455X (gfx1250) — compile-verified
//
#include <hip/hip_runtime.h>
#include <hip/hip_bf16.h>

#define B_    4
#define P_    48000
#define C_    64
#define NX_   432
#define NY_   496
#define NYNX  (NY_ * NX_)          // 214272
#define TILES_X (NX_ / 16)         // 27
#define BN_EPS 1e-3f

typedef __attribute__((ext_vector_type(2))) float v2f;
typedef __attribute__((ext_vector_type(8))) float v8f;

// ---------------------------------------------------------------------------
// Kernel 1: zero the spatial half (channels 0..63) of each batch's output.
// 219 MB of stores -> use float4 (b128) stores, exact coverage, no tail.
// ---------------------------------------------------------------------------
__global__ __launch_bounds__(256) void zero_spatial_kernel(float4* __restrict__ out4) {
    const int per_batch_zero4 = C_ * NYNX / 4;     // 3,428,352
    const int per_batch_out4  = 128 * NYNX / 4;    // 6,856,704
    int i = blockIdx.x * 256 + threadIdx.x;        // grid sized exactly
    int b   = i / per_batch_zero4;
    int off = i - b * per_batch_zero4;
    out4[(size_t)b * per_batch_out4 + off] = make_float4(0.f, 0.f, 0.f, 0.f);
}

// ---------------------------------------------------------------------------
// Kernel 2: scatter pillar features into the NCHW grid.
// 64 consecutive threads handle one pillar -> coalesced feature reads.
// ---------------------------------------------------------------------------
__global__ __launch_bounds__(256) void scatter_kernel(
    const float* __restrict__ pf, const int* __restrict__ coords,
    float* __restrict__ out) {
    int tid = blockIdx.x * 256 + threadIdx.x;      // exact: P*C / 256 blocks
    int p = tid >> 6;
    int c = tid & 63;
    int b = coords[p * 4 + 0];
    int y = coords[p * 4 + 2];
    int x = coords[p * 4 + 3];
    out[(((size_t)b * 128 + c) * NY_ + y) * NX_ + x] = pf[tid];
}

// ---------------------------------------------------------------------------
// Kernel 3: conv(1->64, 3x3, SAME) + BN + ReLU as a WMMA f32 GEMM.
// Block = 128 threads = 4 waves; wave g computes channels [16g, 16g+16) for a
// 16-pixel row tile. K = 9 taps padded to 12 -> 3x V_WMMA_F32_16X16X4_F32.
// ---------------------------------------------------------------------------
__global__ __launch_bounds__(128) void conv_bn_relu_wmma_kernel(
    const float* __restrict__ obs,   const float* __restrict__ w,
    const float* __restrict__ gamma, const float* __restrict__ beta,
    const float* __restrict__ mean,  const float* __restrict__ var,
    float* __restrict__ out) {
    const int t    = blockIdx.x;               // tile id over B*NY*TILES_X
    const int g    = threadIdx.x >> 5;         // channel group 0..3
    const int lane = threadIdx.x & 31;
    const int half = lane >> 4;                // 0: lanes 0-15, 1: lanes 16-31
    const int lid  = lane & 15;

    const int b   = t / (NY_ * TILES_X);
    const int rem = t - b * (NY_ * TILES_X);
    const int y   = rem / TILES_X;
    const int x0  = (rem - y * TILES_X) * 16;

    const float* obsb = obs + (size_t)b * NYNX;
    const int ch = g * 16 + lid;               // channel this lane loads A for

    v8f acc = {};
    #pragma unroll
    for (int j = 0; j < 3; ++j) {
        const int kk = j * 4 + 2 * half;       // K index of first element

        // --- A fragment: weights W[ch][kk], W[ch][kk+1] (0 beyond tap 8) ---
        v2f a;
        a.x = (kk     < 9) ? w[ch * 9 + kk]     : 0.f;
        a.y = (kk + 1 < 9) ? w[ch * 9 + kk + 1] : 0.f;

        // --- B fragment: patch[kk][x0+lid], patch[kk+1][x0+lid] -----------
        float bv[2];
        #pragma unroll
        for (int v = 0; v < 2; ++v) {
            const int k = kk + v;
            const int dy = k / 3 - 1;
            const int dx = k - (k / 3) * 3 - 1;
            const int yy = y + dy;
            const int xx = x0 + lid + dx;
            // branchless: clamp address, select result (keeps EXEC all-1s)
            const bool ok = ((unsigned)yy < (unsigned)NY_) &&
                            ((unsigned)xx < (unsigned)NX_) && (k < 9);
            const int yc = ok ? yy : 0;
            const int xc = ok ? xx : 0;
            const float l = obsb[yc * NX_ + xc];
            bv[v] = ok ? l : 0.f;
        }
        v2f bfrag;
        bfrag.x = bv[0];
        bfrag.y = bv[1];

        // D = A(16x4) * B(4x16) + C   (8 args: neg_a, A, neg_b, B, c_mod, C, reuse_a, reuse_b)
        acc = __builtin_amdgcn_wmma_f32_16x16x4_f32(
            false, a, false, bfrag, (short)0, acc, false, false);
    }

    // --- BN + ReLU + store: d[v] = D[M = v + 8*half][N = lid] -------------
    const size_t outbase = ((size_t)b * 128 + 64) * NYNX;  // obs_feat channels 64..127
    #pragma unroll
    for (int v = 0; v < 8; ++v) {
        const int c = g * 16 + v + 8 * half;
        const float scale = gamma[c] / sqrtf(var[c] + BN_EPS);
        const float shift = beta[c] - mean[c] * scale;
        float r = acc[v] * scale + shift;
        r = r > 0.f ? r : 0.f;
        out[outbase + (size_t)c * NYNX + (size_t)y * NX_ + x0 + lid] = r;
    }
}

// ---------------------------------------------------------------------------
extern "C" void kernel_launch(void* const* d_in, const int* in_sizes, int n_in,
                              void* d_out, int out_size, void* d_ws, size_t ws_size,
                              hipStream_t stream) {
    const float* pf     = (const float*)d_in[0];   // (P, 64)
    const int*   coords = (const int*)  d_in[1];   // (P, 4)
    const float* obs    = (const float*)d_in[2];   // (B, NY, NX)
    const float* conv_w = (const float*)d_in[3];   // (64, 1, 3, 3)
    const float* gamma  = (const float*)d_in[4];
    const float* beta   = (const float*)d_in[5];
    const float* mean   = (const float*)d_in[6];
    const float* var    = (const float*)d_in[7];
    float* out = (float*)d_out;                    // (B, 128, NY, NX)

    // 1) zero spatial half: B*64*NYNX/4 float4s = 13,713,408 -> 53568 blocks
    {
        const int total4 = B_ * C_ * NYNX / 4;
        zero_spatial_kernel<<<total4 / 256, 256, 0, stream>>>((float4*)out);
    }
    // 2) scatter pillar features: P*C = 3,072,000 -> 12000 blocks
    {
        const int total = P_ * C_;
        scatter_kernel<<<total / 256, 256, 0, stream>>>(pf, coords, out);
    }
    // 3) conv + BN + ReLU via WMMA: one block per 16-pixel tile
    {
        const int tiles = B_ * NY_ * TILES_X;      // 53568
        conv_bn_relu_wmma_kernel<<<tiles, 128, 0, stream>>>(
            obs, conv_w, gamma, beta, mean, var, out);
    }
}